// TripletJSLoss_90872918049294
// MI455X (gfx1250) — compile-verified
//
#include <hip/hip_runtime.h>
#include <hip/hip_bf16.h>
#include <math.h>

typedef __attribute__((ext_vector_type(2))) float v2f;
typedef __attribute__((ext_vector_type(8))) float v8f;

#define Bn 64
#define Dn 256
#define Sn 100
#define Kaug 512          // 2*Dn: [x^2 | x]
#define ROWS (Bn * Sn)    // 6400
#define LOG2PI 1.8378770664093453f
#define LOG2F_ 0.6931471805599453f
#define MARGIN 0.3f

// ---------------------------------------------------------------------------
// K1: per-distribution weights: Waug[j][d] = 1/var, Waug[j][D+d] = -2*mu/var
//     cvec[j] = sum_d (mu^2/var + log var) + D*log(2pi)
// ---------------------------------------------------------------------------
__global__ void prep_w(const float* __restrict__ feats,
                       const float* __restrict__ cov,
                       float* __restrict__ Waug,
                       float* __restrict__ cvec) {
    __shared__ float sm[Dn];
    const int j = blockIdx.x;
    const int d = threadIdx.x;
    const float mu = feats[j * Dn + d];
    const float v  = cov[j * Dn + d];
    const float inv = 1.0f / v;
    Waug[(size_t)j * Kaug + d]      = inv;
    Waug[(size_t)j * Kaug + Dn + d] = -2.0f * mu * inv;
    sm[d] = mu * mu * inv + logf(v);
    __syncthreads();
    for (int off = Dn / 2; off > 0; off >>= 1) {
        if (d < off) sm[d] += sm[d + off];
        __syncthreads();
    }
    if (d == 0) cvec[j] = sm[0] + (float)Dn * LOG2PI;
}

// ---------------------------------------------------------------------------
// K2: augmented samples: x = mu + sqrt(var)*eps ; Xaug[row] = [x^2 | x]
//     row = i*S + s ; one block per row, one thread per d
// ---------------------------------------------------------------------------
__global__ void prep_x(const float* __restrict__ feats,
                       const float* __restrict__ cov,
                       const float* __restrict__ eps,
                       float* __restrict__ Xaug) {
    const int row = blockIdx.x;       // 0..6399
    const int d   = threadIdx.x;      // 0..255
    const int i   = row / Sn;
    const float mu = feats[i * Dn + d];
    const float sd = sqrtf(cov[i * Dn + d]);
    const float x  = fmaf(sd, eps[(size_t)row * Dn + d], mu);
    Xaug[(size_t)row * Kaug + d]      = x * x;
    Xaug[(size_t)row * Kaug + Dn + d] = x;
}

// ---------------------------------------------------------------------------
// K3: FP32 WMMA GEMM: lp[row, j] = -0.5 * (Xaug[row] . Waug[j] + cvec[j])
//     Each wave computes a 16-row strip x all 64 columns (4 accumulators),
//     looping K=512 in steps of 4 with V_WMMA_F32_16X16X4_F32.
//     Block = 128 threads = 4 waves, each wave owns one m-tile.
// ---------------------------------------------------------------------------
__global__ void gemm_lp(const float* __restrict__ Xaug,
                        const float* __restrict__ Waug,
                        const float* __restrict__ cvec,
                        float* __restrict__ lp) {
    const int wave = threadIdx.x >> 5;
    const int lane = threadIdx.x & 31;
    const int mtile = blockIdx.x * 4 + wave;       // 0..399
    const int m0 = mtile * 16;
    const int lr = lane & 15;                      // row-in-tile for A, col for B
    const int koff = (lane >> 4) * 2;              // K sub-offset per half-wave

    v8f acc0 = {}, acc1 = {}, acc2 = {}, acc3 = {};

    const float* xrow = Xaug + (size_t)(m0 + lr) * Kaug + koff;
    const float* w0 = Waug + (size_t)(0 * 16 + lr) * Kaug + koff;
    const float* w1 = Waug + (size_t)(1 * 16 + lr) * Kaug + koff;
    const float* w2 = Waug + (size_t)(2 * 16 + lr) * Kaug + koff;
    const float* w3 = Waug + (size_t)(3 * 16 + lr) * Kaug + koff;

#pragma unroll 4
    for (int k = 0; k < Kaug; k += 4) {
        const v2f a  = *(const v2f*)(xrow + k);   // A frag: M=lr, K={k+koff, k+koff+1}
        const v2f b0 = *(const v2f*)(w0 + k);     // B frag: N=lr, same K pair
        const v2f b1 = *(const v2f*)(w1 + k);
        const v2f b2 = *(const v2f*)(w2 + k);
        const v2f b3 = *(const v2f*)(w3 + k);
        acc0 = __builtin_amdgcn_wmma_f32_16x16x4_f32(false, a, false, b0, (short)0, acc0, false, false);
        acc1 = __builtin_amdgcn_wmma_f32_16x16x4_f32(false, a, false, b1, (short)0, acc1, false, false);
        acc2 = __builtin_amdgcn_wmma_f32_16x16x4_f32(false, a, false, b2, (short)0, acc2, false, false);
        acc3 = __builtin_amdgcn_wmma_f32_16x16x4_f32(false, a, false, b3, (short)0, acc3, false, false);
    }

    // C/D layout: VGPR r -> M = m0 + r + 8*(lane>=16), N = lane&15 (+16*g)
    const int rowbase = m0 + ((lane >> 4) << 3);
    v8f accs[4] = {acc0, acc1, acc2, acc3};
#pragma unroll
    for (int g = 0; g < 4; ++g) {
        const int col = g * 16 + lr;
        const float c = cvec[col];
#pragma unroll
        for (int r = 0; r < 8; ++r) {
            lp[(size_t)(rowbase + r) * Bn + col] = -0.5f * (accs[g][r] + c);
        }
    }
}

// ---------------------------------------------------------------------------
// K4: A[i,j] = mean_s ( lp_self - (logaddexp(lp_self, lp) - log2) )
//     lp_self[i,s] = lp[(i*S+s), i]
// ---------------------------------------------------------------------------
__global__ void jsd_accum(const float* __restrict__ lp,
                          float* __restrict__ Amat) {
    __shared__ float sm[128];
    const int i = blockIdx.x;
    const int j = blockIdx.y;
    float sum = 0.0f;
    for (int s = threadIdx.x; s < Sn; s += blockDim.x) {
        const size_t rb = (size_t)(i * Sn + s) * Bn;
        const float a = lp[rb + i];   // log p_i(x_i_s)
        const float b = lp[rb + j];   // log p_j(x_i_s)
        const float mx = fmaxf(a, b);
        const float mn = fminf(a, b);
        const float lae = mx + log1pf(expf(mn - mx));   // logaddexp(a,b)
        sum += a - lae + LOG2F_;
    }
    sm[threadIdx.x] = sum;
    __syncthreads();
    for (int off = 64; off > 0; off >>= 1) {
        if (threadIdx.x < off) sm[threadIdx.x] += sm[threadIdx.x + off];
        __syncthreads();
    }
    if (threadIdx.x == 0) Amat[i * Bn + j] = sm[0] * (1.0f / (float)Sn);
}

// ---------------------------------------------------------------------------
// K5: symmetrize, zero diagonal, hard mining, mean margin-ReLU -> scalar loss
// ---------------------------------------------------------------------------
__global__ void triplet_loss(const float* __restrict__ Amat,
                             const int* __restrict__ targets,
                             float* __restrict__ out) {
    __shared__ float sm[Bn];
    const int i = threadIdx.x;        // 64 threads
    const int ti = targets[i];
    float ap = -INFINITY;
    float an = INFINITY;
    for (int j = 0; j < Bn; ++j) {
        float dist = (i == j) ? 0.0f : 0.5f * (Amat[i * Bn + j] + Amat[j * Bn + i]);
        if (targets[j] == ti) ap = fmaxf(ap, dist);
        else                  an = fminf(an, dist);
    }
    sm[i] = fmaxf(ap - an + MARGIN, 0.0f);
    __syncthreads();
    for (int off = Bn / 2; off > 0; off >>= 1) {
        if (i < off) sm[i] += sm[i + off];
        __syncthreads();
    }
    if (i == 0) out[0] = sm[0] * (1.0f / (float)Bn);   // LOSS_WEIGHT = 1
}

// ---------------------------------------------------------------------------
extern "C" void kernel_launch(void* const* d_in, const int* in_sizes, int n_in,
                              void* d_out, int out_size, void* d_ws, size_t ws_size,
                              hipStream_t stream) {
    const float* feats   = (const float*)d_in[0];   // (64, 256)
    const float* cov     = (const float*)d_in[1];   // (64, 256)
    const float* eps     = (const float*)d_in[2];   // (64, 100, 256)
    const int*   targets = (const int*)d_in[3];     // (64,)
    float* out = (float*)d_out;

    // workspace layout (floats)
    float* Xaug = (float*)d_ws;                          // 6400*512
    float* Waug = Xaug + (size_t)ROWS * Kaug;            // 64*512
    float* cvec = Waug + (size_t)Bn * Kaug;              // 64
    float* lp   = cvec + Bn;                             // 6400*64
    float* Amat = lp + (size_t)ROWS * Bn;                // 64*64

    prep_w<<<Bn, Dn, 0, stream>>>(feats, cov, Waug, cvec);
    prep_x<<<ROWS, Dn, 0, stream>>>(feats, cov, eps, Xaug);
    gemm_lp<<<ROWS / 16 / 4, 128, 0, stream>>>(Xaug, Waug, cvec, lp);
    jsd_accum<<<dim3(Bn, Bn), 128, 0, stream>>>(lp, Amat);
    triplet_loss<<<1, Bn, 0, stream>>>(Amat, targets, out);
}